// DownSample_7945689498135
// MI455X (gfx1250) — compile-verified
//
#include <hip/hip_runtime.h>
#include <hip/hip_bf16.h>
#include <math.h>

// ---------------------------------------------------------------------------
// Shapes (from reference): bs=64, sp_emb=256, dn_in=128, dn_out=256,
// coor_emb=32, N_STK=64, N_STK_PNT=64, n_half=32, out_width=32.
// Output = concat(sparse_out[64*256*32], dense_out[64*256*1024],
//                 stk_coor_sampled[64*32*32])  (all f32)
// ---------------------------------------------------------------------------

typedef __attribute__((ext_vector_type(2))) float v2f;
typedef __attribute__((ext_vector_type(8))) float v8f;
typedef __attribute__((ext_vector_type(4))) unsigned int u32x4;
typedef __attribute__((ext_vector_type(8))) int i32x8;
typedef __attribute__((ext_vector_type(4))) int i32x4;

// LDS X layout: [68-float zeroed prefix][128 rows of 68 floats (64 data + 4 pad)]
// TDM pad_enable inserts the 4-dword pad after every 64 dwords; pads are
// pre-zeroed so pos = -1 (conv left padding) resolves to a zero without
// branches (row ic, pos -1 -> pad word 67 of row ic-1; ic==0 -> prefix).
#define XROW 68

// ---------------------------------------------------------------------------
// 1) Farthest point sampling: one block per batch, one thread per point.
//    Also writes stk_coor_sampled directly.
// ---------------------------------------------------------------------------
__global__ __launch_bounds__(64) void fps_kernel(const float* __restrict__ xyz,
                                                 int* __restrict__ fps_idx,
                                                 float* __restrict__ coor_out) {
    __shared__ float cent[32];
    __shared__ float rd[64];
    __shared__ int   ri[64];
    __shared__ int   s_far;

    const int b   = blockIdx.x;
    const int tid = threadIdx.x;
    const float* base = xyz + (size_t)b * 64 * 32;

    float pt[32];
#pragma unroll
    for (int c = 0; c < 32; ++c) pt[c] = base[tid * 32 + c];

    float dist = 1e10f;
    if (tid == 0) s_far = 0;
    __syncthreads();

    for (int i = 0; i < 32; ++i) {
        const int far = s_far;
        __syncthreads();
        if (tid < 32) cent[tid] = base[far * 32 + tid];
        __syncthreads();

        if (tid == 0) fps_idx[b * 32 + i] = far;
        if (tid < 32) coor_out[((size_t)b * 32 + i) * 32 + tid] = cent[tid];

        float d = 0.0f;
#pragma unroll
        for (int c = 0; c < 32; ++c) { float df = pt[c] - cent[c]; d += df * df; }
        dist = fminf(dist, d);

        rd[tid] = dist; ri[tid] = tid;
        __syncthreads();
        // argmax, first-index tie-break (matches jnp.argmax)
        for (int s = 32; s > 0; s >>= 1) {
            if (tid < s) {
                float a = rd[tid], c2 = rd[tid + s];
                int   ia = ri[tid], ib = ri[tid + s];
                if (c2 > a || (c2 == a && ib < ia)) { rd[tid] = c2; ri[tid] = ib; }
            }
            __syncthreads();
        }
        if (tid == 0) s_far = ri[0];
        __syncthreads();
    }
}

// ---------------------------------------------------------------------------
// 2) Repack conv_w [256,128,1,3] into per-lane WMMA A-fragment order:
//    Wr[((t*32+kb)*16 + mt)*64 + lane*2 + v] = W[oc, ic, t]
//    with oc = mt*16 + lane%16, ic = kb*4 + (lane/16)*2 + v
//    -> A-fragment load becomes one coalesced 64-bit load per lane.
// ---------------------------------------------------------------------------
__global__ void repack_w_kernel(const float* __restrict__ w, float* __restrict__ wr) {
    const int idx = blockIdx.x * blockDim.x + threadIdx.x;   // < 98304
    const int v    = idx & 1;
    const int lane = (idx >> 1) & 31;
    const int mt   = (idx >> 6) & 15;
    const int kb   = (idx >> 10) & 31;
    const int t    = idx >> 15;
    const int oc = mt * 16 + (lane & 15);
    const int ic = kb * 4 + (lane >> 4) * 2 + v;
    wr[idx] = w[oc * 384 + ic * 3 + t];
}

// ---------------------------------------------------------------------------
// 3) Gather sparse features of sampled strokes: [64,256,64] -> [64,256,32]
// ---------------------------------------------------------------------------
__global__ void sparse_gather_kernel(const float* __restrict__ sp,
                                     const int* __restrict__ fps_idx,
                                     float* __restrict__ out) {
    const int i = blockIdx.x * blockDim.x + threadIdx.x;     // < 524288
    const int j = i & 31;
    const int e = (i >> 5) & 255;
    const int b = i >> 13;
    out[i] = sp[((size_t)b * 256 + e) * 64 + fps_idx[b * 32 + j]];
}

// ---------------------------------------------------------------------------
// 4) Conv (1x3, stride 2, pad 1) as WMMA GEMM. One block per (batch, stroke):
//    Y[256 oc, 32 ow] = sum_t W_t[256,128] @ X_t[128,32].
//    128 threads = 4 waves; each wave: 4 M-tiles x 2 N-tiles, K=4 f32 WMMA.
//    X tile (128 ic x 64 pos, 32KB) staged LDS-side by the Tensor Data Mover
//    in one tensor_load_to_lds (row pad 64->68 via D# pad fields).
// ---------------------------------------------------------------------------
__global__ __launch_bounds__(128) void conv_wmma_kernel(
        const float* __restrict__ dense, const int* __restrict__ fps_idx,
        const float* __restrict__ Wr, const float* __restrict__ conv_b,
        float* __restrict__ y) {
    __shared__ float Xs[XROW + 128 * XROW];   // 35,088 bytes

    const int blk  = blockIdx.x;
    const int b    = blk >> 5;
    const int j    = blk & 31;
    const int tid  = threadIdx.x;
    const int wave = tid >> 5;
    const int lane = tid & 31;
    const int stroke = fps_idx[b * 32 + j];

    const float* src = dense + (size_t)b * 128 * 4096 + stroke * 64;

    // zero the prefix row and the 4 pad words of every row (TDM skips pads)
    for (int q = tid; q < XROW; q += 128) Xs[q] = 0.0f;
    {
        const float4 z = {0.0f, 0.0f, 0.0f, 0.0f};
        *(float4*)&Xs[XROW + tid * XROW + 64] = z;   // tid = row 0..127
    }

#if __has_builtin(__builtin_amdgcn_tensor_load_to_lds)
    if (wave == 0) {
        // ---- Tensor DMA descriptor (D#) ----
        u32x4 g0;
        g0[0] = 1u;                                   // count=1, user mode
        g0[1] = (unsigned)(size_t)(void*)&Xs[XROW];   // lds_addr (byte offset)
        const unsigned long long ga = (unsigned long long)(size_t)(const void*)src;
        g0[2] = (unsigned)ga;                         // global_addr[31:0]
        g0[3] = ((unsigned)(ga >> 32) & 0x01FFFFFFu) | (2u << 30);  // addr hi | type=2

        i32x8 g1;
        g1[0] = (2 << 16)      // data_size = 4 bytes
              | (1 << 20)      // pad_enable
              | (5 << 22)      // pad_interval: every 64 dwords
              | (3 << 25);     // pad_amount: 4 dwords
        g1[1] = (64 << 16);            // tensor_dim0[15:0] = 64 (atomic barrier addr = 0)
        g1[2] = (0) | (128 << 16);     // tensor_dim0 hi = 0, tensor_dim1[15:0] = 128
        g1[3] = (0) | (64 << 16);      // tensor_dim1 hi = 0, tile_dim0 = 64
        g1[4] = (128) | (0 << 16);     // tile_dim1 = 128, tile_dim2 = 0 (2D)
        g1[5] = 4096;                  // tensor_dim0_stride[31:0]
        g1[6] = 0;                     // stride hi, tensor_dim1_stride lo (unused)
        g1[7] = 0;

        const i32x4 z4 = {0, 0, 0, 0};
#if __has_include(<hip/amd_detail/amd_gfx1250_TDM.h>)
        const i32x8 z8 = {0, 0, 0, 0, 0, 0, 0, 0};
        __builtin_amdgcn_tensor_load_to_lds(g0, g1, z4, z4, z8, 0);
#else
        __builtin_amdgcn_tensor_load_to_lds(g0, g1, z4, z4, 0);
#endif
        __builtin_amdgcn_s_wait_tensorcnt(0);
    }
#else
    // Fallback: manual staging ([ic][pos] layout needs no transpose)
    for (int l = tid; l < 2048; l += 128) {
        const int ic = l >> 4;
        const int p0 = (l & 15) << 2;
        const float4 v = *(const float4*)(src + (size_t)ic * 4096 + p0);
        *(float4*)&Xs[XROW + ic * XROW + p0] = v;
    }
#endif
    __syncthreads();

    v8f acc[4][2];
#pragma unroll
    for (int mt = 0; mt < 4; ++mt)
#pragma unroll
        for (int nt = 0; nt < 2; ++nt)
            acc[mt][nt] = (v8f){0.f, 0.f, 0.f, 0.f, 0.f, 0.f, 0.f, 0.f};

    const int half = lane >> 4;   // 0: K 0..1 / M 0..7 half, 1: K 2..3 / M 8..15
    const int lrow = lane & 15;
    const float* xrow = &Xs[XROW];

    for (int t = 0; t < 3; ++t) {
        const int pos0 = 2 * lrow - 1 + t;        // nt=0 column (may be -1 -> zero pad)
        const float* wrt = Wr + (size_t)t * 32768 + wave * 256 + lane * 2;
#pragma unroll 4
        for (int kb = 0; kb < 32; ++kb) {
            const int icA = kb * 4 + half * 2;
            const float* xr0 = xrow + icA * XROW;
            const v2f bf0 = {xr0[pos0],        xr0[XROW + pos0]};
            const v2f bf1 = {xr0[pos0 + 32],   xr0[XROW + pos0 + 32]};
            const float* wk = wrt + kb * 1024;
#pragma unroll
            for (int mt = 0; mt < 4; ++mt) {
                const v2f af = *(const v2f*)(wk + mt * 64);
                acc[mt][0] = __builtin_amdgcn_wmma_f32_16x16x4_f32(
                    false, af, false, bf0, (short)0, acc[mt][0], false, false);
                acc[mt][1] = __builtin_amdgcn_wmma_f32_16x16x4_f32(
                    false, af, false, bf1, (short)0, acc[mt][1], false, false);
            }
        }
    }

    // D layout: lane -> col N = lane%16, vgpr v -> row M = v + 8*(lane/16)
#pragma unroll
    for (int mt = 0; mt < 4; ++mt) {
#pragma unroll
        for (int v = 0; v < 8; ++v) {
            const int oc = wave * 64 + mt * 16 + half * 8 + v;
            const float bias = conv_b[oc];
            const size_t base = (((size_t)b * 256 + oc) * 32 + j) * 32;
            y[base + lrow]      = acc[mt][0][v] + bias;
            y[base + 16 + lrow] = acc[mt][1][v] + bias;
        }
    }
}

// ---------------------------------------------------------------------------
// 5) BatchNorm batch statistics: one block per channel; writes mean & rstd
//    (no atomics -> no zero-init of workspace needed).
// ---------------------------------------------------------------------------
__global__ __launch_bounds__(256) void stats_kernel(const float* __restrict__ y,
                                                    float* __restrict__ stats) {
    __shared__ float s1[256];
    __shared__ float s2[256];
    const int oc = blockIdx.x;
    const int tid = threadIdx.x;
    float sum = 0.0f, sq = 0.0f;
    for (int b = 0; b < 64; ++b) {
        const float* p = y + ((size_t)b * 256 + oc) * 1024;
        for (int k = tid; k < 1024; k += 256) {
            const float v = p[k];
            sum += v; sq += v * v;
        }
    }
    s1[tid] = sum; s2[tid] = sq;
    __syncthreads();
    for (int s = 128; s > 0; s >>= 1) {
        if (tid < s) { s1[tid] += s1[tid + s]; s2[tid] += s2[tid + s]; }
        __syncthreads();
    }
    if (tid == 0) {
        const float mean = s1[0] * (1.0f / 65536.0f);
        const float var  = s2[0] * (1.0f / 65536.0f) - mean * mean;
        stats[oc]       = mean;
        stats[256 + oc] = rsqrtf(var + 1e-5f);
    }
}

// ---------------------------------------------------------------------------
// 6) BN normalize + tanh-GELU, in place over the dense output region.
// ---------------------------------------------------------------------------
__global__ __launch_bounds__(256) void bn_gelu_kernel(float* __restrict__ y,
                                                      const float* __restrict__ stats,
                                                      const float* __restrict__ gamma,
                                                      const float* __restrict__ beta) {
    const size_t i = (size_t)blockIdx.x * 256 + threadIdx.x;   // < 16777216
    const int oc = (int)((i >> 10) & 255);
    float v = y[i];
    v = (v - stats[oc]) * stats[256 + oc] * gamma[oc] + beta[oc];
    const float t = tanhf(0.7978845608028654f * (v + 0.044715f * v * v * v));
    y[i] = 0.5f * v * (1.0f + t);
}

// ---------------------------------------------------------------------------
extern "C" void kernel_launch(void* const* d_in, const int* in_sizes, int n_in,
                              void* d_out, int out_size, void* d_ws, size_t ws_size,
                              hipStream_t stream) {
    const float* sparse_fea = (const float*)d_in[0];   // [64,256,64]
    const float* dense_fea  = (const float*)d_in[1];   // [64,128,4096]
    const float* stk_coor   = (const float*)d_in[2];   // [64,64,32]
    const float* conv_w     = (const float*)d_in[3];   // [256,128,1,3]
    const float* conv_b     = (const float*)d_in[4];   // [256]
    const float* bn_gamma   = (const float*)d_in[5];   // [256]
    const float* bn_beta    = (const float*)d_in[6];   // [256]

    float* out        = (float*)d_out;
    float* out_sparse = out;                            // 524288
    float* out_dense  = out + 524288;                   // 16777216
    float* out_coor   = out + 524288 + 16777216;        // 65536

    int*   fps_ws = (int*)d_ws;                                         // 2048 ints
    float* Wr     = (float*)((char*)d_ws + 16384);                      // 98304 f32
    float* stats  = (float*)((char*)d_ws + 16384 + 98304 * 4);          // 512 f32

    fps_kernel<<<64, 64, 0, stream>>>(stk_coor, fps_ws, out_coor);
    repack_w_kernel<<<384, 256, 0, stream>>>(conv_w, Wr);
    sparse_gather_kernel<<<2048, 256, 0, stream>>>(sparse_fea, fps_ws, out_sparse);
    conv_wmma_kernel<<<2048, 128, 0, stream>>>(dense_fea, fps_ws, Wr, conv_b, out_dense);
    stats_kernel<<<256, 256, 0, stream>>>(out_dense, stats);
    bn_gelu_kernel<<<65536, 256, 0, stream>>>(out_dense, stats, bn_gamma, bn_beta);
}